// LocalNormalizedCrossCorrelationDEBUG1_15710990369527
// MI455X (gfx1250) — compile-verified
//
#include <hip/hip_runtime.h>

// LNCC DEBUG1: num = (box(t*p) - box(p)/729 * box(t))^2 + 1e-5, 9^3 box, SAME zero pad.
// Strategy (MI455X, 23.3 TB/s HBM, 192 MB L2):
//   Pass1: streaming 9-tap running sums along D (stride 192^2) from raw inputs,
//          producing t,p,tp fields in d_ws (coalesced along W).
//   Pass2: same along H (stride 192), in place (per-thread column: read j before write j-4).
//   Pass3: along W via WMMA banded matmul: Out(16x16) = A(data 16x4) x B(band 4x16),
//          <=6 K-blocks per tile, fused final combine. v_wmma_f32_16x16x4_f32.
//   Effective HBM traffic ~340 MB (intermediates are L2-resident) -> ~15 us.

typedef __attribute__((ext_vector_type(2))) float v2f;
typedef __attribute__((ext_vector_type(8))) float v8f;

#define DIMB 2
#define DIM 192
#define SLICE (DIM * DIM)         // 36864
#define VOL (DIM * SLICE)         // 7077888
#define NELEM (DIMB * VOL)        // 14155776
#define INV_VOL (1.0f / 729.0f)
#define EPS_LNCC 1e-5f

// ---------------- Pass 1: filter along D, build t/p/tp fields ----------------
__global__ __launch_bounds__(256) void lncc_pass_d(const float* __restrict__ yt,
                                                   const float* __restrict__ yp,
                                                   float* __restrict__ t,
                                                   float* __restrict__ p,
                                                   float* __restrict__ q) {
  int col = blockIdx.x * 256 + threadIdx.x;       // (b, h, w) column: 2*192*192 = 73728
  int b = col / SLICE;
  int hw = col - b * SLICE;
  int base = b * VOL + hw;                        // stride along D = SLICE

  float rt[9], rp[9], rq[9];
#pragma unroll
  for (int i = 0; i < 9; ++i) { rt[i] = 0.f; rp[i] = 0.f; rq[i] = 0.f; }
  float st = 0.f, sp = 0.f, sq = 0.f;

  int j = 0;
  for (int m = 0; m < 22; ++m) {                  // 22*9 = 198 >= 192+4+1 steps
#pragma unroll
    for (int u = 0; u < 9; ++u) {                 // j % 9 == u  (static ring index)
      float vt = 0.f, vp = 0.f, vq = 0.f;
      if (j < DIM) {
        int idx = base + j * SLICE;
        vt = yt[idx];
        vp = yp[idx];
        vq = vt * vp;
      }
      st += vt - rt[u]; rt[u] = vt;               // window [j-8, j] == [c-4, c+4]
      sp += vp - rp[u]; rp[u] = vp;
      sq += vq - rq[u]; rq[u] = vq;
      int c = j - 4;
      if (c >= 0 && c < DIM) {
        int odx = base + c * SLICE;
        t[odx] = st; p[odx] = sp; q[odx] = sq;
      }
      ++j;
    }
  }
}

// ---------------- Pass 2: filter along H, in place on the fields ----------------
__global__ __launch_bounds__(256) void lncc_pass_h(float* __restrict__ t,
                                                   float* __restrict__ p,
                                                   float* __restrict__ q) {
  int col = blockIdx.x * 256 + threadIdx.x;       // (b, d, w) column: 73728
  int b = col / SLICE;
  int rem = col - b * SLICE;
  int d = rem / DIM;
  int w = rem - d * DIM;
  int base = b * VOL + d * SLICE + w;             // stride along H = DIM

  float rt[9], rp[9], rq[9];
#pragma unroll
  for (int i = 0; i < 9; ++i) { rt[i] = 0.f; rp[i] = 0.f; rq[i] = 0.f; }
  float st = 0.f, sp = 0.f, sq = 0.f;

  int j = 0;
  for (int m = 0; m < 22; ++m) {
#pragma unroll
    for (int u = 0; u < 9; ++u) {
      float vt = 0.f, vp = 0.f, vq = 0.f;
      if (j < DIM) {
        int idx = base + j * DIM;
        vt = t[idx]; vp = p[idx]; vq = q[idx];
      }
      st += vt - rt[u]; rt[u] = vt;
      sp += vp - rp[u]; rp[u] = vp;
      sq += vq - rq[u]; rq[u] = vq;
      int c = j - 4;
      if (c >= 0 && c < DIM) {
        int odx = base + c * DIM;                 // in-place: c < j, read-before-write per thread
        t[odx] = st; p[odx] = sp; q[odx] = sq;
      }
      ++j;
    }
  }
}

// ---------------- Pass 3: filter along W via WMMA band-matmul + combine ----------------
// One wave per 16(h) x 16(w) output tile of a (b,d) slice.
// Out = A(16x4 data) x B(4x16 band of 0/1) accumulated over K-blocks; K spans [16T-4, 16T+19].
// Wave-uniform control (slice/tile/loop bounds) is forced into SGPRs via readfirstlane so the
// K-block loop is a scalar-branch loop: EXEC is never modified around v_wmma (ISA requirement).
__global__ __launch_bounds__(256) void lncc_pass_w_wmma(const float* __restrict__ t,
                                                        const float* __restrict__ p,
                                                        const float* __restrict__ q,
                                                        float* __restrict__ out) {
  // wave id is uniform within the wave; readfirstlane pins it (and all derived
  // control values) to SGPRs -> scalar loop, full-EXEC WMMA.
  int wave = __builtin_amdgcn_readfirstlane(blockIdx.x * 8 + (threadIdx.x >> 5));
  int lane = threadIdx.x & 31;

  int slice = wave / 144;                           // (b*192 + d), 384 slices
  int tt = wave - slice * 144;
  int rowTile = tt / 12;                            // h tile
  int colTile = tt - rowTile * 12;                  // w tile
  int sbase = slice * SLICE;

  int m = lane & 15;                                // A: M = lane%16
  int kHalf = (lane >> 4) * 2;                      // A/B: K pair = (lane/16)*2 + {0,1}
  int rowA = rowTile * 16 + m;
  int n = colTile * 16 + m;                         // B/C: N = lane%16

  v8f ct = {}, cp = {}, cq = {};

  int kb0 = 4 * colTile - 1; if (kb0 < 0) kb0 = 0;  // clip = implicit zero padding
  int kb1 = 4 * colTile + 4; if (kb1 > 47) kb1 = 47;

  for (int kb = kb0; kb <= kb1; ++kb) {             // scalar (SGPR) loop: EXEC untouched
    int x = 4 * kb + kHalf;
    int aoff = sbase + rowA * DIM + x;              // 8B-aligned (x even)
    v2f at = *(const v2f*)(t + aoff);               // A tile: global_load_b64 per lane
    v2f ap = *(const v2f*)(p + aoff);
    v2f aq = *(const v2f*)(q + aoff);

    v2f bb;                                         // band: 1 if |x_k - n| <= 4
    int d0 = x - n;
    int d1 = d0 + 1;
    bb.x = (d0 >= -4 && d0 <= 4) ? 1.f : 0.f;
    bb.y = (d1 >= -4 && d1 <= 4) ? 1.f : 0.f;

    ct = __builtin_amdgcn_wmma_f32_16x16x4_f32(false, at, false, bb, (short)0, ct, false, false);
    cp = __builtin_amdgcn_wmma_f32_16x16x4_f32(false, ap, false, bb, (short)0, cp, false, false);
    cq = __builtin_amdgcn_wmma_f32_16x16x4_f32(false, aq, false, bb, (short)0, cq, false, false);
  }

  int xo = colTile * 16 + m;
  int rhalf = (lane >> 4) * 8;                      // C/D: VGPR v -> M = v + 8*(lane/16)
#pragma unroll
  for (int v = 0; v < 8; ++v) {
    int row = rowTile * 16 + v + rhalf;
    float ts = ct[v], ps = cp[v], qs = cq[v];
    float cross = qs - (ps * INV_VOL) * ts;
    out[sbase + row * DIM + xo] = cross * cross + EPS_LNCC;
  }
}

extern "C" void kernel_launch(void* const* d_in, const int* in_sizes, int n_in,
                              void* d_out, int out_size, void* d_ws, size_t ws_size,
                              hipStream_t stream) {
  const float* yt = (const float*)d_in[0];
  const float* yp = (const float*)d_in[1];
  float* out = (float*)d_out;

  size_t need = (size_t)3 * (size_t)NELEM * sizeof(float);   // ~162 MiB, fits in 192 MB L2
  if (ws_size < need) return;                                // safety: avoid corrupting memory

  float* t = (float*)d_ws;
  float* p = t + NELEM;
  float* q = p + NELEM;

  lncc_pass_d<<<288, 256, 0, stream>>>(yt, yp, t, p, q);     // 73728 threads exactly
  lncc_pass_h<<<288, 256, 0, stream>>>(t, p, q);             // in-place H filter
  lncc_pass_w_wmma<<<6912, 256, 0, stream>>>(t, p, q, out);  // 55296 waves exactly
}